// PointCloudModel_83176336654880
// MI455X (gfx1250) — compile-verified
//
#include <hip/hip_runtime.h>

typedef __attribute__((ext_vector_type(2))) float v2f;
typedef __attribute__((ext_vector_type(8))) float v8f;
typedef __attribute__((ext_vector_type(4))) int   v4i;
typedef __attribute__((ext_vector_type(4))) float v4f;

#define HF 64   // hidden feature width

// ================= common small kernels =================

// dinv = deg^-0.5
__global__ void dinv_kernel(const float* __restrict__ cnt, float* __restrict__ dinv, int n) {
    int i = blockIdx.x * blockDim.x + threadIdx.x;
    if (i < n) dinv[i] = rsqrtf(cnt[i]);
}

// ================= CSR (gather) path =================

// cnt_row = 1.0 (self-loop), cnt_col = 1 (self-loop)
__global__ void init_csr_kernel(float* __restrict__ cnt_row, int* __restrict__ cnt_col, int n) {
    int i = blockIdx.x * blockDim.x + threadIdx.x;
    if (i < n) { cnt_row[i] = 1.0f; cnt_col[i] = 1; }
}

// out-degree over row (for dinv) and in-degree over col (for CSR) in one pass
__global__ void count_kernel(const int* __restrict__ row, const int* __restrict__ col,
                             float* __restrict__ cnt_row, int* __restrict__ cnt_col, int E) {
    int e = blockIdx.x * blockDim.x + threadIdx.x;
    if (e < E) {
        int r = __builtin_nontemporal_load(row + e);
        int c = __builtin_nontemporal_load(col + e);
        atomicAdd(&cnt_row[r], 1.0f);
        atomicAdd(&cnt_col[c], 1);
    }
}

// single-block exclusive scan: indptr[i] = sum(cnt[0..i)), indptr[n] = total;
// wcnt gets a copy (bin-write cursors). 256 threads, LDS Hillis-Steele per chunk.
__global__ void scan_kernel(const int* __restrict__ cnt, int* __restrict__ indptr,
                            int* __restrict__ wcnt, int n) {
    __shared__ int sdata[256];
    __shared__ int srun;
    int tid = threadIdx.x;
    if (tid == 0) srun = 0;
    __syncthreads();
    for (int base = 0; base < n; base += 256) {
        int i = base + tid;
        int v = (i < n) ? cnt[i] : 0;
        sdata[tid] = v;
        __syncthreads();
#pragma unroll
        for (int off = 1; off < 256; off <<= 1) {
            int t = (tid >= off) ? sdata[tid - off] : 0;
            __syncthreads();
            sdata[tid] += t;
            __syncthreads();
        }
        int incl = sdata[tid];
        int excl = incl - v;
        int run  = srun;                       // stable: last write pre-synced
        if (i < n) { indptr[i] = run + excl; wcnt[i] = run + excl; }
        __syncthreads();
        if (tid == 255) srun = run + incl;     // chunk total
        __syncthreads();
    }
    if (tid == 0) indptr[n] = srun;
}

// bin edges by destination: srcbin[indptr[c] ...] = list of sources of c
__global__ void bin_kernel(const int* __restrict__ row, const int* __restrict__ col,
                           int* __restrict__ wcnt, int* __restrict__ srcbin, int E, int Etot) {
    int e = blockIdx.x * blockDim.x + threadIdx.x;
    if (e >= Etot) return;
    int r, c;
    if (e < E) {
        r = __builtin_nontemporal_load(row + e);
        c = __builtin_nontemporal_load(col + e);
    } else {
        r = e - E; c = r;                      // appended self-loops
    }
    int pos = atomicAdd(&wcnt[c], 1);
    __builtin_nontemporal_store(r, srcbin + pos);
}

// gather aggregation: one wave per node, 2 features/lane (b64 loads), no atomics.
// agg[i] = dinv[i] * sum_s dinv[s] * h[s]   over CSR sources s of node i.
__global__ void gather_kernel(const int* __restrict__ indptr, const int* __restrict__ srcbin,
                              const float* __restrict__ h, const float* __restrict__ dinv,
                              float* __restrict__ agg, int n) {
    int wid  = (blockIdx.x * blockDim.x + threadIdx.x) >> 5;   // wave32
    int lane = threadIdx.x & 31;
    if (wid >= n) return;
    int start = indptr[wid];
    int end   = indptr[wid + 1];
    v2f acc; acc.x = 0.0f; acc.y = 0.0f;
    int fo = lane * 2;
    for (int j = start; j < end; ++j) {
        int s   = srcbin[j];                   // same addr across wave -> broadcast
        float w = dinv[s];
        v2f hv  = *(const v2f*)(h + (size_t)s * HF + fo);      // coalesced 256B row
        acc.x += w * hv.x;
        acc.y += w * hv.y;
    }
    float di = dinv[wid];
    acc.x *= di; acc.y *= di;
    *(v2f*)(agg + (size_t)wid * HF + fo) = acc;
}

// ================= atomic (fallback) path =================

__global__ void init_atomic_kernel(v4f* __restrict__ agg4, float* __restrict__ cnt, int n) {
    int i = blockIdx.x * blockDim.x + threadIdx.x;
    if (i < n * (HF / 4)) agg4[i] = (v4f){0.f, 0.f, 0.f, 0.f};
    if (i < n) cnt[i] = 1.0f;
}

__global__ void degree_kernel(const int* __restrict__ row, float* __restrict__ cnt, int E) {
    int e = blockIdx.x * blockDim.x + threadIdx.x;
    if (e < E) atomicAdd(&cnt[__builtin_nontemporal_load(row + e)], 1.0f);
}

// half-warp per edge: one b128 gather + 4 atomic f32 adds
__global__ void scatter_kernel(const int* __restrict__ row, const int* __restrict__ col,
                               const float* __restrict__ h, const float* __restrict__ dinv,
                               float* __restrict__ agg, int E, int Etot) {
    int gid = blockIdx.x * blockDim.x + threadIdx.x;
    int e   = gid >> 4;
    int sl  = gid & 15;
    if (e >= Etot) return;
    int r, c;
    if (e < E) {
        r = __builtin_nontemporal_load(row + e);
        c = __builtin_nontemporal_load(col + e);
    } else {
        r = e - E; c = r;
    }
    float norm = dinv[r] * dinv[c];
    v4f hv = *((const v4f*)(h + (size_t)r * HF) + sl);
    float* ac = agg + (size_t)c * HF + sl * 4;
    atomicAdd(ac + 0, norm * hv.x);
    atomicAdd(ac + 1, norm * hv.y);
    atomicAdd(ac + 2, norm * hv.z);
    atomicAdd(ac + 3, norm * hv.w);
}

// ================= WMMA GEMMs =================

// GEMM1: h[n,64] = x[n,6] @ W1[64,6]^T + b1  (f32 16x16x4, K padded 6->8)
__global__ void gemm1_wmma(const float* __restrict__ x, const float* __restrict__ W1,
                           const float* __restrict__ b1, float* __restrict__ h, int n) {
    int wave = threadIdx.x >> 5;
    int lane = threadIdx.x & 31;
    int lm   = lane & 15;
    bool hi  = lane >= 16;
    int m0   = blockIdx.x * 16;
    int j0   = wave * 16;
    int koff = hi ? 2 : 0;

    int m = m0 + lm; if (m >= n) m = n - 1;
    const float* xr = x + (size_t)m * 6;
    v2f a0, a1;
    a0.x = xr[koff]; a0.y = xr[koff + 1];
    float x4 = xr[4], x5 = xr[5];
    a1.x = hi ? 0.0f : x4;
    a1.y = hi ? 0.0f : x5;

    int j = j0 + lm;
    const float* wr = W1 + (size_t)j * 6;
    v2f b0v, b1v;
    b0v.x = wr[koff]; b0v.y = wr[koff + 1];
    float w4 = wr[4], w5 = wr[5];
    b1v.x = hi ? 0.0f : w4;
    b1v.y = hi ? 0.0f : w5;

    float bias = b1[j];
    v8f c;
#pragma unroll
    for (int v = 0; v < 8; ++v) c[v] = bias;

    c = __builtin_amdgcn_wmma_f32_16x16x4_f32(false, a0, false, b0v, (short)0, c, false, false);
    c = __builtin_amdgcn_wmma_f32_16x16x4_f32(false, a1, false, b1v, (short)0, c, false, false);

    int mrow0 = m0 + (hi ? 8 : 0);
    float* hp = h + (size_t)mrow0 * HF + j0 + lm;
    if (m0 + 16 <= n) {
#pragma unroll
        for (int v = 0; v < 8; ++v) hp[(size_t)v * HF] = c[v];
    } else {
#pragma unroll
        for (int v = 0; v < 8; ++v)
            if (mrow0 + v < n) hp[(size_t)v * HF] = c[v];
    }
}

// GEMM2: out[n,3] = relu(agg[n,64]) @ W2[3,64]^T + b2  (16 K-steps of 16x16x4)
__global__ void gemm2_wmma(const float* __restrict__ agg, const float* __restrict__ W2,
                           const float* __restrict__ b2, float* __restrict__ out, int n) {
    int wave = threadIdx.x >> 5;
    int lane = threadIdx.x & 31;
    int tile = blockIdx.x * 8 + wave;
    int m0   = tile * 16;
    if (m0 >= n) return;
    int lm  = lane & 15;
    bool hi = lane >= 16;

    int m = m0 + lm; if (m >= n) m = n - 1;
    const float* ar = agg + (size_t)m * HF + (hi ? 2 : 0);

    int j  = lm;
    int jc = j < 3 ? j : 2;
    bool jv = j < 3;
    const float* wr = W2 + (size_t)jc * HF + (hi ? 2 : 0);

    float bias = jv ? b2[jc] : 0.0f;
    v8f c;
#pragma unroll
    for (int v = 0; v < 8; ++v) c[v] = bias;

#pragma unroll
    for (int s = 0; s < 16; ++s) {
        int k0 = 4 * s;
        v2f a, b;
        a.x = fmaxf(ar[k0],     0.0f);
        a.y = fmaxf(ar[k0 + 1], 0.0f);
        float wv0 = wr[k0], wv1 = wr[k0 + 1];
        b.x = jv ? wv0 : 0.0f;
        b.y = jv ? wv1 : 0.0f;
        c = __builtin_amdgcn_wmma_f32_16x16x4_f32(false, a, false, b, (short)0, c, false, false);
    }

    int mrow0 = m0 + (hi ? 8 : 0);
    if (lm < 3) {
        float* op = out + (size_t)mrow0 * 3 + lm;
        if (m0 + 16 <= n) {
#pragma unroll
            for (int v = 0; v < 8; ++v) op[(size_t)v * 3] = c[v];
        } else {
#pragma unroll
            for (int v = 0; v < 8; ++v)
                if (mrow0 + v < n) op[(size_t)v * 3] = c[v];
        }
    }
}

// ================= launcher =================

extern "C" void kernel_launch(void* const* d_in, const int* in_sizes, int n_in,
                              void* d_out, int out_size, void* d_ws, size_t ws_size,
                              hipStream_t stream) {
    const float* x   = (const float*)d_in[0];
    const int*   ei  = (const int*)d_in[1];   // [2, E] int32
    const float* W1  = (const float*)d_in[2]; // [64,6]
    const float* b1  = (const float*)d_in[3]; // [64]
    const float* W2  = (const float*)d_in[4]; // [3,64]
    const float* b2  = (const float*)d_in[5]; // [3]
    float* out = (float*)d_out;

    int n = in_sizes[0] / 6;
    int E = in_sizes[1] / 2;
    int Etot = E + n;
    const int* row  = ei;
    const int* colp = ei + E;

    const int T = 256;
    int ntiles = (n + 15) / 16;

    // CSR layout: h | agg | dinv | cnt_row | cnt_col | indptr | wcnt | srcbin
    size_t csrElems = (size_t)2 * n * HF + (size_t)n * 4 + 1 + (size_t)Etot;
    bool useCSR = ws_size >= csrElems * sizeof(float) + 64;

    float* h    = (float*)d_ws;                     // n*64
    float* agg  = h + (size_t)n * HF;               // n*64
    float* dinv = agg + (size_t)n * HF;             // n

    if (useCSR) {
        float* cnt_row = dinv + n;                  // n
        int*   cnt_col = (int*)(cnt_row + n);       // n
        int*   indptr  = cnt_col + n;               // n+1
        int*   wcnt    = indptr + (n + 1);          // n
        int*   srcbin  = wcnt + n;                  // Etot

        init_csr_kernel<<<(n + T - 1) / T, T, 0, stream>>>(cnt_row, cnt_col, n);
        count_kernel<<<(E + T - 1) / T, T, 0, stream>>>(row, colp, cnt_row, cnt_col, E);
        dinv_kernel<<<(n + T - 1) / T, T, 0, stream>>>(cnt_row, dinv, n);
        scan_kernel<<<1, 256, 0, stream>>>(cnt_col, indptr, wcnt, n);
        bin_kernel<<<(Etot + T - 1) / T, T, 0, stream>>>(row, colp, wcnt, srcbin, E, Etot);
        gemm1_wmma<<<ntiles, 128, 0, stream>>>(x, W1, b1, h, n);
        long long gthreads = (long long)n * 32;
        gather_kernel<<<(unsigned)((gthreads + T - 1) / T), T, 0, stream>>>(
            indptr, srcbin, h, dinv, agg, n);
    } else {
        float* cnt_row = dinv + n;                  // n
        unsigned gInit = (unsigned)(((size_t)n * (HF / 4) + T - 1) / T);
        init_atomic_kernel<<<gInit, T, 0, stream>>>((v4f*)agg, cnt_row, n);
        degree_kernel<<<(E + T - 1) / T, T, 0, stream>>>(row, cnt_row, E);
        dinv_kernel<<<(n + T - 1) / T, T, 0, stream>>>(cnt_row, dinv, n);
        gemm1_wmma<<<ntiles, 128, 0, stream>>>(x, W1, b1, h, n);
        long long sthreads = (long long)Etot * 16;
        scatter_kernel<<<(unsigned)((sthreads + T - 1) / T), T, 0, stream>>>(
            row, colp, h, dinv, agg, E, Etot);
    }

    gemm2_wmma<<<(ntiles + 7) / 8, 256, 0, stream>>>(agg, W2, b2, out, n);
}